// FTScanBasic_56298431316003
// MI455X (gfx1250) — compile-verified
//
#include <hip/hip_runtime.h>
#include <hip/hip_bf16.h>

// Problem shape (fixed by the reference).
#define T_ROWS   8192
#define D_COLS   4096
#define CHUNK    128                 // rows per chunk (8 WMMA tiles of 16)
#define NCHUNK   (T_ROWS / CHUNK)    // 64
#define TILES_PER_CHUNK (CHUNK / 16) // 8

typedef __attribute__((ext_vector_type(2))) float v2f;
typedef __attribute__((ext_vector_type(8))) float v8f;

// ---------------------------------------------------------------------------
// Phase 1: per-(column, chunk) partial column sums. Coalesced streaming read;
// pulls the whole 128 MiB input into the 192 MB L2 for phase 3's re-read.
// ---------------------------------------------------------------------------
__global__ void scan_phase1_chunksum(const float* __restrict__ xs,
                                     float* __restrict__ partial) {
    const int col   = blockIdx.x * blockDim.x + threadIdx.x; // 0..4095
    const int chunk = blockIdx.y;                            // 0..63
    const float* p = xs + (size_t)chunk * CHUNK * D_COLS + col;
    float s = 0.0f;
#pragma unroll 8
    for (int r = 0; r < CHUNK; ++r) {
        s += p[(size_t)r * D_COLS];
    }
    partial[chunk * D_COLS + col] = s;
}

// ---------------------------------------------------------------------------
// Phase 2: per-column exclusive scan of the 64 chunk sums (in place) and the
// final carry (= total column sum) written to d_out[0:4096].
// ---------------------------------------------------------------------------
__global__ void scan_phase2_prefix(float* __restrict__ partial,
                                   float* __restrict__ final_carry) {
    const int col = blockIdx.x * blockDim.x + threadIdx.x; // 0..4095
    float run = 0.0f;
    for (int chunk = 0; chunk < NCHUNK; ++chunk) {
        const float t = partial[chunk * D_COLS + col];
        partial[chunk * D_COLS + col] = run;   // exclusive prefix for phase 3
        run += t;
    }
    __builtin_nontemporal_store(run, &final_carry[col]);
}

// ---------------------------------------------------------------------------
// Phase 3: one wave per (16-column tile, 128-row chunk). For each 16x16 tile:
//   Y = L * X + C,  L = lower-triangular ones, C = carry broadcast per column
// realized as 4 chained V_WMMA_F32_16X16X4_F32 (K split 16 = 4x4).
//
// ISA 16x4 f32 A layout:  VGPR0 = K {0 | 2}, VGPR1 = K {1 | 3}; lanes 0-15
// hold M=lane, lanes 16-31 hold M=lane-16 with K offset +2. B (4x16) mirrors
// this per-row-in-lane layout; C/D: VGPR r = rows {r | r+8} across lane halves.
// ---------------------------------------------------------------------------
__global__ void scan_phase3_wmma(const float* __restrict__ xs,
                                 const float* __restrict__ partial,
                                 float* __restrict__ ys) {
    const int lane  = threadIdx.x & 31;
    const int wave  = threadIdx.x >> 5;                  // 0..3
    const int colTile = blockIdx.x * 4 + wave;           // 0..255
    const int c0    = colTile * 16;
    const int chunk = blockIdx.y;                        // 0..63
    const int r0    = chunk * CHUNK;
    const int half  = lane >> 4;                         // 0 or 1
    const int n     = lane & 15;                         // column within tile

    // Lower-triangular ones, laid out as four 16x4 A tiles (loop-invariant).
    v2f a[4];
#pragma unroll
    for (int t = 0; t < 4; ++t) {
        const int k0 = 4 * t + 2 * half;     // VGPR0 K index for this lane half
        a[t].x = (k0     <= n) ? 1.0f : 0.0f;  // M = n
        a[t].y = (k0 + 1 <= n) ? 1.0f : 0.0f;
    }

    // Incoming carry for this chunk (exclusive prefix of chunk sums).
    float carry = partial[chunk * D_COLS + c0 + n];

    for (int tile = 0; tile < TILES_PER_CHUNK; ++tile) {
        const int rbase = r0 + tile * 16;

        // C preloaded with carry broadcast down every row of the tile.
        v8f c = {carry, carry, carry, carry, carry, carry, carry, carry};

#pragma unroll
        for (int t = 0; t < 4; ++t) {
            const int k0 = 4 * t + 2 * half;
            v2f b;
            b.x = xs[(size_t)(rbase + k0)     * D_COLS + c0 + n];
            b.y = xs[(size_t)(rbase + k0 + 1) * D_COLS + c0 + n];
            // D = A*B + C ; f32 in / f32 accumulate (exact match to reference)
            c = __builtin_amdgcn_wmma_f32_16x16x4_f32(
                    /*neg_a=*/false, a[t], /*neg_b=*/false, b,
                    /*c_mod=*/(short)0, c, /*reuse_a=*/false, /*reuse_b=*/false);
        }

        // Store the scanned tile: VGPR r -> rows rbase+r (lanes 0-15) and
        // rbase+r+8 (lanes 16-31). Non-temporal: don't evict input from L2.
#pragma unroll
        for (int r = 0; r < 8; ++r) {
            __builtin_nontemporal_store(
                c[r], &ys[(size_t)(rbase + r + 8 * half) * D_COLS + c0 + n]);
        }

        // New carry = row 15 of D = c[7] on lanes 16..31; broadcast so that
        // every lane holds the carry of its own column n.
        carry = __shfl(c[7], 16 + n, 32);
    }
}

// ---------------------------------------------------------------------------
extern "C" void kernel_launch(void* const* d_in, const int* in_sizes, int n_in,
                              void* d_out, int out_size, void* d_ws, size_t ws_size,
                              hipStream_t stream) {
    const float* xs = (const float*)d_in[0];
    float* out         = (float*)d_out;
    float* final_carry = out;            // [4096]
    float* ys          = out + D_COLS;   // [8192, 4096]
    float* partial     = (float*)d_ws;   // [NCHUNK, D_COLS] = 1 MiB

    // Phase 1: chunk column sums.
    scan_phase1_chunksum<<<dim3(D_COLS / 256, NCHUNK), 256, 0, stream>>>(xs, partial);

    // Phase 2: exclusive scan of chunk sums + final carry.
    scan_phase2_prefix<<<dim3(D_COLS / 256), 256, 0, stream>>>(partial, final_carry);

    // Phase 3: WMMA triangular-matmul scan within each chunk.
    // 4 waves/block, each wave owns one 16-column tile; grid = (256/4, 64).
    scan_phase3_wmma<<<dim3(64, NCHUNK), 128, 0, stream>>>(xs, partial, ys);
}